// SOMLayer_62165356642732
// MI455X (gfx1250) — compile-verified
//
#include <hip/hip_runtime.h>

typedef float v2f __attribute__((ext_vector_type(2)));
typedef float v8f __attribute__((ext_vector_type(8)));

#define OH 125
#define OW 125
#define P_TOT (OH * OW)      // 15625 patches per batch
#define DDIM 256             // C*KH*KW
#define KCODE 256            // H*W codebook entries
#define CIN 16
#define HIN 128
#define WIN 128
#define NTILES_M ((P_TOT + 15) / 16)   // 977

// ---------------------------------------------------------------------------
// Kernel 1: f2[k] = sum_d som[k*DDIM + d]^2   (K = 256, D = 256; trivial)
// ---------------------------------------------------------------------------
__global__ void som_f2_kernel(const float* __restrict__ som,
                              float* __restrict__ f2) {
    int k = blockIdx.x * blockDim.x + threadIdx.x;
    if (k >= KCODE) return;
    const float* row = som + (size_t)k * DDIM;
    float s = 0.0f;
#pragma unroll 8
    for (int d = 0; d < DDIM; ++d) s = fmaf(row[d], row[d], s);
    f2[k] = s;
}

// ---------------------------------------------------------------------------
// Kernel 2: per-wave WMMA GEMM + argmin.
// One wave handles 16 consecutive patches (M-tile) x all 256 codebooks.
// score[m][n] = f2[n] + sum_d (-2*pat[m][d]) * f[n][d]   (exact f32, WMMA)
// argmin over n with first-index tie-break -> winner (row, col) / 16.
// ---------------------------------------------------------------------------
__global__ __launch_bounds__(256)
void som_argmin_kernel(const float* __restrict__ x,
                       const float* __restrict__ som,
                       const float* __restrict__ f2,
                       float* __restrict__ out) {
    const int lane = threadIdx.x & 31;
    const int wave = threadIdx.x >> 5;
    const int b    = blockIdx.y;
    const int tile = blockIdx.x * 8 + wave;
    if (tile >= NTILES_M) return;            // whole wave exits; others have EXEC all-1s

    const int m  = lane & 15;                // patch-in-tile (A rows) / codebook-in-group (B cols)
    const int hi = lane >> 4;                // selects d-pair {0,1} vs {2,3} within k=4 step

    int p = tile * 16 + m;
    if (p >= P_TOT) p = P_TOT - 1;           // clamp loads for tail tile (writes guarded later)
    const int oh = p / OW;
    const int ow = p - oh * OW;

    // Per-lane base pointers; everything else is a compile-time immediate offset.
    // A (patches): d = 16*c + 4*kh + kw ; step kk covers d = 4*kk .. 4*kk+3
    //   lane half hi: kw = 2*hi + {0,1}; offset(kk) = (kk>>2)*HIN*WIN + (kk&3)*WIN
    const float* xbase = x + (size_t)b * CIN * HIN * WIN + oh * WIN + ow + hi * 2;
    // B (codebook, f^T): n = m, rows d = 4*kk + 2*hi + {0,1}; nt adds 16 codebooks
    const float* fbase = som + (size_t)m * DDIM + hi * 2;
    const float* f2base = f2 + m;

    // f2 cache: one value per codebook-group
    float f2c[16];
#pragma unroll
    for (int nt = 0; nt < 16; ++nt) f2c[nt] = f2base[nt * 16];

    // 16 independent accumulators (all 256 codebooks), init C = f2[n]
    v8f acc[16];
#pragma unroll
    for (int nt = 0; nt < 16; ++nt) {
        v8f c;
#pragma unroll
        for (int j = 0; j < 8; ++j) c[j] = f2c[nt];
        acc[nt] = c;
    }

    // Main K loop: 64 steps of k=4.  A loaded once per step, scaled by -2
    // (exact); 16 independent WMMAs per step -> no D->A/B RAW hazards.
#pragma unroll 2
    for (int kk = 0; kk < 64; ++kk) {
        const float* xa = xbase + (kk >> 2) * (HIN * WIN) + (kk & 3) * WIN;
        v2f a;
        a.x = xa[0] * -2.0f;
        a.y = xa[1] * -2.0f;
#pragma unroll
        for (int nt = 0; nt < 16; ++nt) {
            const float* fb = fbase + nt * (16 * DDIM) + kk * 4;
            v2f bf = *(const v2f*)fb;        // 8B-aligned by construction
            acc[nt] = __builtin_amdgcn_wmma_f32_16x16x4_f32(
                false, a, false, bf, (short)0, acc[nt], false, false);
        }
    }

    // Per-lane argmin over the 16 codebook groups (ascending idx => first-min)
    float bestv[8];
    int   besti[8];
#pragma unroll
    for (int j = 0; j < 8; ++j) { bestv[j] = __builtin_inff(); besti[j] = 0x7fffffff; }
#pragma unroll
    for (int nt = 0; nt < 16; ++nt) {
        const int idx = nt * 16 + m;
#pragma unroll
        for (int j = 0; j < 8; ++j) {
            float v = acc[nt][j];
            if (v < bestv[j] || (v == bestv[j] && idx < besti[j])) {
                bestv[j] = v; besti[j] = idx;
            }
        }
    }

    // Cross-lane reduction within each 16-lane half (xor masks < 16 never mix
    // halves).  C layout: VGPR j holds M=j (lanes 0-15) and M=j+8 (lanes 16-31).
#pragma unroll
    for (int j = 0; j < 8; ++j) {
        float v = bestv[j];
        int idx = besti[j];
#pragma unroll
        for (int off = 1; off < 16; off <<= 1) {
            float ov = __shfl_xor(v, off, 32);
            int   oi = __shfl_xor(idx, off, 32);
            if (ov < v || (ov == v && oi < idx)) { v = ov; idx = oi; }
        }
        if (m == 0) {
            const int mm = j + hi * 8;
            const int pp = tile * 16 + mm;
            if (pp < P_TOT) {
                const float wr = (float)(idx >> 4) * 0.0625f;   // (idx/W)/H
                const float wc = (float)(idx & 15) * 0.0625f;   // (idx%W)/W
                out[(size_t)b * 2 * P_TOT + pp]         = wr;
                out[(size_t)b * 2 * P_TOT + P_TOT + pp] = wc;
            }
        }
    }
}

// ---------------------------------------------------------------------------
extern "C" void kernel_launch(void* const* d_in, const int* in_sizes, int n_in,
                              void* d_out, int out_size, void* d_ws, size_t ws_size,
                              hipStream_t stream) {
    const float* x   = (const float*)d_in[0];   // (16,16,128,128) f32
    const float* som = (const float*)d_in[1];   // (16,16,16,4,4) f32 == (256,256)
    float* f2  = (float*)d_ws;                  // 256 floats of scratch
    float* out = (float*)d_out;                 // (16,2,125,125) f32

    som_f2_kernel<<<dim3(1), dim3(256), 0, stream>>>(som, f2);

    dim3 grid((NTILES_M + 7) / 8, 16, 1);       // 8 waves (M-tiles) per block, y = batch
    som_argmin_kernel<<<grid, dim3(256), 0, stream>>>(x, som, f2, out);
}